// attbase_53764400611553
// MI455X (gfx1250) — compile-verified
//
#include <hip/hip_runtime.h>
#include <cstdint>

// Reference reduces exactly to a depthwise 3x3 conv (pad 1) of x with wv:
// softmax over a size-1 axis == 1.0, so the attention path is the identity.
// Pure bandwidth-bound kernel: stage each 128x128 plane into LDS via CDNA5
// async global->LDS copies, compute 3x3 taps from LDS, stream out with NT.

typedef float v4f __attribute__((ext_vector_type(4)));

#define C_   256
#define H_   128
#define W_   128
#define TSTRIDE 136   // floats per LDS tile row (544 B: 16B aligned, no pad-col aliasing)
#define TROWS   130   // 128 rows + top/bottom halo

__global__ __launch_bounds__(256) void attbase_dwconv3x3_kernel(
    const float* __restrict__ x,
    const float* __restrict__ wv,
    float* __restrict__ out)
{
    __shared__ __align__(16) float tile[TROWS * TSTRIDE]; // 70,720 B

    const int tid   = threadIdx.x;
    const int plane = blockIdx.x;                 // b*256 + c
    const int c     = plane & (C_ - 1);
    const size_t pbase = (size_t)plane * (H_ * W_);
    const float* src = x + pbase;
    float* dst = out + pbase;

    // ---- zero only the halo (disjoint from async-written interior) ----
    if (tid < 130) {
        tile[0   * TSTRIDE + 3 + tid] = 0.0f;     // row above tensor row 0
        tile[129 * TSTRIDE + 3 + tid] = 0.0f;     // row below tensor row 127
    }
    if (tid < 128) {
        tile[(tid + 1) * TSTRIDE + 3]   = 0.0f;   // col left of col 0
        tile[(tid + 1) * TSTRIDE + 132] = 0.0f;   // col right of col 127
    }

    // ---- async copy plane (64 KiB) into interior: rows 1..128, cols 4..131 ----
    // Tensor row r, vec4 j -> LDS byte offset base + ((r+1)*136 + 4 + 4j)*4
    // = base + (r+1)*544 + 16 + 16j  (16B aligned for B128 LDS writes).
    const unsigned ldsBase = (unsigned)(uintptr_t)(&tile[0]);
    #pragma unroll
    for (int it = 0; it < 16; ++it) {
        int v = it * 256 + tid;                   // vec4 id 0..4095
        int r = v >> 5;                           // row 0..127
        int j = v & 31;                           // vec4 column 0..31
        unsigned ldsOff = ldsBase + (unsigned)(((r + 1) * TSTRIDE + 4 + j * 4) * 4);
        unsigned long long ga =
            (unsigned long long)(uintptr_t)(src + r * W_ + j * 4);
        asm volatile("global_load_async_to_lds_b128 %0, %1, off"
                     :: "v"(ldsOff), "v"(ga) : "memory");
    }
    asm volatile("s_wait_asynccnt 0x0" ::: "memory");
    __syncthreads();

    // ---- per-channel 3x3 weights (cross-correlation, as XLA conv) ----
    const float* wp = wv + c * 9;
    const float w00 = wp[0], w01 = wp[1], w02 = wp[2];
    const float w10 = wp[3], w11 = wp[4], w12 = wp[5];
    const float w20 = wp[6], w21 = wp[7], w22 = wp[8];

    // ---- compute: 16 iterations; per iteration each wave does one full row,
    //      each lane 4 consecutive columns (coalesced 512B NT stores/wave) ----
    const int lane = tid & 31;
    const int wid  = tid >> 5;                    // wave id 0..7
    for (int it = 0; it < 16; ++it) {
        int r  = it * 8 + wid;                    // output row 0..127
        int c4 = lane * 4;                        // output col base 0..124
        int i0 = r * TSTRIDE + 4 + c4;            // tile row r   == tensor row r-1
        int i1 = i0 + TSTRIDE;                    // tensor row r
        int i2 = i1 + TSTRIDE;                    // tensor row r+1

        v4f A = *(const v4f*)&tile[i0]; float L0 = tile[i0 - 1], R0 = tile[i0 + 4];
        v4f B = *(const v4f*)&tile[i1]; float L1 = tile[i1 - 1], R1 = tile[i1 + 4];
        v4f D = *(const v4f*)&tile[i2]; float L2 = tile[i2 - 1], R2 = tile[i2 + 4];

        v4f o;
        o.x = w00*L0  + w01*A.x + w02*A.y
            + w10*L1  + w11*B.x + w12*B.y
            + w20*L2  + w21*D.x + w22*D.y;
        o.y = w00*A.x + w01*A.y + w02*A.z
            + w10*B.x + w11*B.y + w12*B.z
            + w20*D.x + w21*D.y + w22*D.z;
        o.z = w00*A.y + w01*A.z + w02*A.w
            + w10*B.y + w11*B.z + w12*B.w
            + w20*D.y + w21*D.z + w22*D.w;
        o.w = w00*A.z + w01*A.w + w02*R0
            + w10*B.z + w11*B.w + w12*R1
            + w20*D.z + w21*D.w + w22*R2;

        __builtin_nontemporal_store(o, (v4f*)&dst[r * W_ + c4]);
    }
}

extern "C" void kernel_launch(void* const* d_in, const int* in_sizes, int n_in,
                              void* d_out, int out_size, void* d_ws, size_t ws_size,
                              hipStream_t stream) {
    const float* x  = (const float*)d_in[0];
    // d_in[1] = wq, d_in[2] = wk: dead code (softmax over size-1 axis == 1.0).
    const float* wv = (const float*)d_in[3];
    float* out = (float*)d_out;

    (void)in_sizes; (void)n_in; (void)out_size; (void)d_ws; (void)ws_size;

    attbase_dwconv3x3_kernel<<<8 * C_, 256, 0, stream>>>(x, wv, out);
}